// DecoderModel_42228118454332
// MI455X (gfx1250) — compile-verified
//
#include <hip/hip_runtime.h>
#include <math.h>

typedef __attribute__((ext_vector_type(2))) float v2f;
typedef __attribute__((ext_vector_type(8))) float v8f;

// ---------------------------------------------------------------------------
// Phase 0: compose the linear (no-activation) alignment MLP into a single
// 2048-float vector w_P (the part of W4@W3@W2@W1 acting on `padded`).
// Biases and the S-dependent part cancel under softmax over t.
// ---------------------------------------------------------------------------
__global__ void align_combine(const float* __restrict__ W1,  // [256,3072]
                              const float* __restrict__ W2,  // [256,256]
                              const float* __restrict__ W3,  // [256,256]
                              const float* __restrict__ W4,  // [1,256]
                              float* __restrict__ wP) {      // [2048]
    __shared__ float r4[256], r3[256], r2[256];
    const int t = threadIdx.x;               // 256 threads
    r4[t] = W4[t];
    __syncthreads();
    float s = 0.f;
    for (int k = 0; k < 256; ++k) s += r4[k] * W3[k * 256 + t];   // r3 = r4 @ W3
    r3[t] = s;
    __syncthreads();
    s = 0.f;
    for (int k = 0; k < 256; ++k) s += r3[k] * W2[k * 256 + t];   // r2 = r3 @ W2
    r2[t] = s;
    __syncthreads();
    for (int j = t; j < 2048; j += 256) {                         // wP = r2 @ W1[:,1024:]
        float acc = 0.f;
        for (int k = 0; k < 256; ++k) acc += r2[k] * W1[k * 3072 + 1024 + j];
        wP[j] = acc;
    }
}

// ---------------------------------------------------------------------------
// Phase 1a: e[t,b] = padded[t,b,:] . wP   (one wave per (t,b); 3200 waves)
// ---------------------------------------------------------------------------
__global__ void attn_scores(const float* __restrict__ padded,  // [50,64,2048]
                            const float* __restrict__ wP,      // [2048]
                            float* __restrict__ e) {           // [50,64]
    const int wave = blockIdx.x * (blockDim.x >> 5) + (threadIdx.x >> 5);
    const int lane = threadIdx.x & 31;
    if (wave >= 50 * 64) return;
    const float* row = padded + (size_t)wave * 2048;
    float s = 0.f;
    for (int j = lane; j < 2048; j += 32) s += row[j] * wP[j];
    for (int off = 16; off > 0; off >>= 1) s += __shfl_xor(s, off, 32);
    if (lane == 0) e[wave] = s;
}

// ---------------------------------------------------------------------------
// Phase 1b: in-place softmax over t (dim 0) per batch column. 64 columns.
// ---------------------------------------------------------------------------
__global__ void attn_softmax(float* __restrict__ e) {  // [50,64]
    const int b = threadIdx.x;
    if (b >= 64) return;
    float mx = -1e30f;
    for (int t = 0; t < 50; ++t) mx = fmaxf(mx, e[t * 64 + b]);
    float sum = 0.f;
    for (int t = 0; t < 50; ++t) {
        const float v = expf(e[t * 64 + b] - mx);
        e[t * 64 + b] = v;
        sum += v;
    }
    const float inv = 1.f / sum;
    for (int t = 0; t < 50; ++t) e[t * 64 + b] *= inv;
}

// ---------------------------------------------------------------------------
// Phase 1c: context[b,j] = sum_t a[t,b] * padded[t,b,j]
// ---------------------------------------------------------------------------
__global__ void attn_context(const float* __restrict__ padded,  // [50,64,2048]
                             const float* __restrict__ a,       // [50,64]
                             float* __restrict__ ctx) {         // [64,2048]
    const int idx = blockIdx.x * blockDim.x + threadIdx.x;      // 131072
    const int b = idx >> 11, j = idx & 2047;
    float s = 0.f;
    for (int t = 0; t < 50; ++t)
        s += a[t * 64 + b] * padded[((size_t)t * 64 + b) * 2048 + j];
    ctx[idx] = s;
}

// ---------------------------------------------------------------------------
// Phase 2 GEMM: G[64,4096] = A[64,K] @ W[4096,K]^T via V_WMMA_F32_16X16X4_F32.
// One wave owns a 16x16 tile. A/B VGPR layout (32-bit, 16x4 / 4x16):
//   lane<16 : K = k0+0 (v0), k0+1 (v1);  lane>=16 : K = k0+2 (v0), k0+3 (v1)
// C/D: VGPR r holds M = r + 8*(lane>=16), N = lane&15.
// ---------------------------------------------------------------------------
__global__ void gemm_wmma_f32(const float* __restrict__ A,  // [64,K]
                              const float* __restrict__ W,  // [4096,K]
                              float* __restrict__ G,        // [64,4096]
                              int K) {
    const int wave = blockIdx.x * (blockDim.x >> 5) + (threadIdx.x >> 5);
    const int lane = threadIdx.x & 31;
    const int mTile = wave >> 8;    // 4 tiles of M
    const int nTile = wave & 255;   // 256 tiles of N
    const int m0 = mTile * 16, n0 = nTile * 16;
    const int half = lane >> 4;     // K sub-pair selector
    const int idx  = lane & 15;     // M (for A) / N (for B) within tile

    const float* aPtr = A + (size_t)(m0 + idx) * K + 2 * half;
    const float* bPtr = W + (size_t)(n0 + idx) * K + 2 * half;

    v8f acc = {};
    for (int k = 0; k < K; k += 4) {
        const v2f a = *(const v2f*)(aPtr + k);
        const v2f b = *(const v2f*)(bPtr + k);
        acc = __builtin_amdgcn_wmma_f32_16x16x4_f32(
            /*neg_a=*/false, a, /*neg_b=*/false, b,
            /*c_mod=*/(short)0, acc, /*reuse_a=*/false, /*reuse_b=*/false);
    }
#pragma unroll
    for (int r = 0; r < 8; ++r) {
        const int m = r + 8 * half;
        G[(size_t)(m0 + m) * 4096 + n0 + idx] = acc[r];
    }
}

// ---------------------------------------------------------------------------
// Phase 2 gates: c = sigmoid(i)*tanh(g); h = sigmoid(o)*tanh(c)  (f gate moot:
// c_prev == 0). Gate order i,f,g,o at offsets 0,1024,2048,3072.
// ---------------------------------------------------------------------------
__global__ void lstm_gates(const float* __restrict__ G,    // [64,4096]
                           const float* __restrict__ bih,  // [4096]
                           const float* __restrict__ bhh,  // [4096]
                           float* __restrict__ h) {        // [64,1024]
    const int idx = blockIdx.x * blockDim.x + threadIdx.x;  // 65536
    const int b = idx >> 10, n = idx & 1023;
    const float* g = G + (size_t)b * 4096;
    const float gi = g[n]        + bih[n]        + bhh[n];
    const float gg = g[2048 + n] + bih[2048 + n] + bhh[2048 + n];
    const float go = g[3072 + n] + bih[3072 + n] + bhh[3072 + n];
    const float c  = (1.f / (1.f + expf(-gi))) * tanhf(gg);
    h[idx]         = (1.f / (1.f + expf(-go))) * tanhf(c);
}

// ---------------------------------------------------------------------------
// Phase 3: every decoder step yields the same h -> broadcast to [50,64,1024].
// ---------------------------------------------------------------------------
__global__ void broadcast_out(const float* __restrict__ h, float* __restrict__ out) {
    const int idx = blockIdx.x * blockDim.x + threadIdx.x;  // 65536
    const float v = h[idx];
    for (int s = 0; s < 50; ++s) out[(size_t)s * 65536 + idx] = v;
}

extern "C" void kernel_launch(void* const* d_in, const int* in_sizes, int n_in,
                              void* d_out, int out_size, void* d_ws, size_t ws_size,
                              hipStream_t stream) {
    // Input order: padded, align_ws[0..3], align_bs[0..3], w_ih[0..3],
    //              w_hh[0..3] (unused, h0=c0=0), b_ih[0..3], b_hh[0..3]
    const float* padded = (const float*)d_in[0];
    const float* aw[4]  = {(const float*)d_in[1], (const float*)d_in[2],
                           (const float*)d_in[3], (const float*)d_in[4]};
    const float* wih[4] = {(const float*)d_in[9],  (const float*)d_in[10],
                           (const float*)d_in[11], (const float*)d_in[12]};
    const float* bih[4] = {(const float*)d_in[17], (const float*)d_in[18],
                           (const float*)d_in[19], (const float*)d_in[20]};
    const float* bhh[4] = {(const float*)d_in[21], (const float*)d_in[22],
                           (const float*)d_in[23], (const float*)d_in[24]};
    float* out = (float*)d_out;

    float* ws  = (float*)d_ws;
    float* wP  = ws;               // 2048
    float* e   = wP + 2048;        // 3200
    float* ctx = e + 3200;         // 64*2048 = 131072
    float* G   = ctx + 131072;     // 64*4096 = 262144
    float* h   = G + 262144;       // 64*1024 = 65536   (total ~1.86 MB)

    // Phase 0: collapse the linear alignment MLP.
    align_combine<<<1, 256, 0, stream>>>(aw[0], aw[1], aw[2], aw[3], wP);

    // Phase 1: step-invariant attention (scores -> softmax over t -> context).
    attn_scores<<<400, 256, 0, stream>>>(padded, wP, e);   // 3200 waves
    attn_softmax<<<1, 64, 0, stream>>>(e);
    attn_context<<<512, 256, 0, stream>>>(padded, e, ctx);

    // Phase 2: single 4-layer LSTM cell (WMMA f32 GEMMs + gate fusion).
    gemm_wmma_f32<<<256, 128, 0, stream>>>(ctx, wih[0], G, 2048);
    lstm_gates<<<256, 256, 0, stream>>>(G, bih[0], bhh[0], h);
    for (int l = 1; l < 4; ++l) {
        gemm_wmma_f32<<<256, 128, 0, stream>>>(h, wih[l], G, 1024);
        lstm_gates<<<256, 256, 0, stream>>>(G, bih[l], bhh[l], h);
    }

    // Phase 3: all 50 steps identical -> broadcast.
    broadcast_out<<<256, 256, 0, stream>>>(h, out);
}